// LinkPredictor_66211215835754
// MI455X (gfx1250) — compile-verified
//
#include <hip/hip_runtime.h>

// GCN link predictor for MI455X (gfx1250, wave32).
// DIM = 64 throughout. fp32 end-to-end; GEMMs use V_WMMA_F32_16X16X4_F32.

#define DIMV 64

typedef __attribute__((ext_vector_type(2))) float v2f;
typedef __attribute__((ext_vector_type(8))) float v8f;

// ---------------------------------------------------------------- utilities

__global__ void init_deg_kernel(float* __restrict__ deg, int n) {
  int i = blockIdx.x * blockDim.x + threadIdx.x;
  if (i < n) deg[i] = 1.0f;  // self loop contributes 1 to every node's degree
}

__global__ void deg_count_kernel(const int* __restrict__ dst, float* deg, int nE) {
  int e = blockIdx.x * blockDim.x + threadIdx.x;
  if (e < nE) atomicAdd(&deg[dst[e]], 1.0f);
}

__global__ void dinv_kernel(const float* __restrict__ deg, float* __restrict__ dinv, int n) {
  int i = blockIdx.x * blockDim.x + threadIdx.x;
  if (i < n) dinv[i] = rsqrtf(deg[i]);  // deg >= 1 always (self loop)
}

__global__ void zero4_kernel(float4* __restrict__ p, int n4) {
  int i = blockIdx.x * blockDim.x + threadIdx.x;
  if (i < n4) p[i] = make_float4(0.f, 0.f, 0.f, 0.f);
}

// ---------------------------------------------------------------- WMMA GEMM
// out[M x 64] = in[M x 64] @ W[64 x 64]   (W row-major: W[k][n])
// Block = 128 threads = 4 waves; each wave computes 16 rows x 64 cols.
// K = 64 -> 16 steps of V_WMMA_F32_16X16X4_F32 per 16x16 tile, 4 tiles/wave.
//
// W is staged into LDS *pre-paired by k*: sW2[p][n] = {W[2p][n], W[2p+1][n]}.
// A lane's B operand (K=kb,kb+1 ; kb = k0+2*hi is always even) is then one
// ds_load_b64 that lands in an even-aligned VGPR pair -> direct WMMA SRC1,
// no repacking movs.

__global__ __launch_bounds__(128) void gemm64_wmma_kernel(
    const float* __restrict__ A, const float* __restrict__ W,
    float* __restrict__ out, int M) {
  __shared__ float2 sW2[(DIMV / 2) * DIMV];  // 16 KB of the 320 KB WGP LDS

  const int wave = threadIdx.x >> 5;
  const int lane = threadIdx.x & 31;
  const int lm   = lane & 15;   // M index within tile (A), N index (B/C/D)
  const int hi   = lane >> 4;   // K-pair select (A/B), M+8 select (C/D)

  const int r0   = blockIdx.x * 64 + wave * 16;
  const int row  = r0 + lm;
  const int rowc = row < M ? row : M - 1;  // clamp loads; stores are guarded
  const float* __restrict__ arow = A + (size_t)rowc * DIMV;

  // Kick the A row (256 B = 2 cachelines) toward L0 while we stage W in LDS.
  __builtin_prefetch(arow, 0, 3);
  __builtin_prefetch(arow + 32, 0, 3);

  // Stage W pre-paired: entry idx=(p,n) holds rows k=2p and k=2p+1, column n.
  for (int idx = threadIdx.x; idx < (DIMV / 2) * DIMV; idx += 128) {
    int p = idx >> 6, n = idx & 63;
    sW2[idx] = make_float2(W[(2 * p) * DIMV + n], W[(2 * p + 1) * DIMV + n]);
  }
  __syncthreads();

  if (r0 >= M) return;  // wave-uniform exit; no barriers after this point

  v8f acc0 = {}, acc1 = {}, acc2 = {}, acc3 = {};

#pragma unroll
  for (int k0 = 0; k0 < DIMV; k0 += 4) {
    const int kb = k0 + 2 * hi;        // this lane's K pair base (even)
    const int p  = kb >> 1;            // paired-row index in sW2
    // A operand (16x4 f32): v0 = K=(0|2), v1 = K=(1|3) per ISA layout.
    v2f a;
    {
      const float2 t = *(const float2*)(arow + kb);
      a.x = t.x; a.y = t.y;
    }
    // B operands (4x16 f32), one aligned 8-byte LDS load per column tile.
    v2f b0, b1, b2, b3;
    { float2 t = sW2[p * DIMV +  0 + lm]; b0.x = t.x; b0.y = t.y; }
    { float2 t = sW2[p * DIMV + 16 + lm]; b1.x = t.x; b1.y = t.y; }
    { float2 t = sW2[p * DIMV + 32 + lm]; b2.x = t.x; b2.y = t.y; }
    { float2 t = sW2[p * DIMV + 48 + lm]; b3.x = t.x; b3.y = t.y; }
    // 8-arg form: (neg_a, A, neg_b, B, c_mod, C, reuse_a, reuse_b)
    acc0 = __builtin_amdgcn_wmma_f32_16x16x4_f32(false, a, false, b0, (short)0, acc0, false, false);
    acc1 = __builtin_amdgcn_wmma_f32_16x16x4_f32(false, a, false, b1, (short)0, acc1, false, false);
    acc2 = __builtin_amdgcn_wmma_f32_16x16x4_f32(false, a, false, b2, (short)0, acc2, false, false);
    acc3 = __builtin_amdgcn_wmma_f32_16x16x4_f32(false, a, false, b3, (short)0, acc3, false, false);
  }

  // C/D layout: VGPR j -> row r0 + j + 8*hi, col = n0 + lm.
#pragma unroll
  for (int j = 0; j < 8; ++j) {
    const int r = r0 + j + 8 * hi;
    if (r < M) {
      float* orow = out + (size_t)r * DIMV;
      orow[ 0 + lm] = acc0[j];
      orow[16 + lm] = acc1[j];
      orow[32 + lm] = acc2[j];
      orow[48 + lm] = acc3[j];
    }
  }
}

// ---------------------------------------------------------------- scatter
// agg[dst[e]][c] += dinv[src[e]]*dinv[dst[e]] * h[src[e]][c]
// One thread per (edge, channel): fully coalesced gathers + atomics; the
// 25.6 MB feature matrix lives in the 192 MB L2, so both sides stay in-cache.

__global__ void scatter_kernel(const float* __restrict__ h,
                               const int* __restrict__ src,
                               const int* __restrict__ dst,
                               const float* __restrict__ dinv,
                               float* agg, int nE) {
  int gid = blockIdx.x * blockDim.x + threadIdx.x;
  int e = gid >> 6;
  int c = gid & 63;
  if (e < nE) {
    int s = src[e], d = dst[e];
    float norm = dinv[s] * dinv[d];
    atomicAdd(&agg[(size_t)d * DIMV + c], norm * h[(size_t)s * DIMV + c]);
  }
}

// ---------------------------------------------------------------- finalize
// agg[i][c] = (relu?)(agg[i][c] + dinv[i]^2 * h[i][c] + b[c])

template <bool RELU>
__global__ void finalize_kernel(float* agg, const float* __restrict__ h,
                                const float* __restrict__ dinv,
                                const float* __restrict__ bias, int n) {
  int gid = blockIdx.x * blockDim.x + threadIdx.x;
  if (gid < n * DIMV) {
    int node = gid >> 6;
    int c = gid & 63;
    float di = dinv[node];
    float v = agg[gid] + di * di * h[gid] + bias[c];
    if (RELU) v = fmaxf(v, 0.0f);
    agg[gid] = v;
  }
}

// ---------------------------------------------------------------- decode
// score[e] = dot(z[src[e]], z[dst[e]]).  16 lanes per edge, float4 loads,
// xor-shuffle tree within the 16-lane group (wave32-safe offsets 8/4/2/1).

__global__ void decode_kernel(const float* __restrict__ z,
                              const int* __restrict__ src,
                              const int* __restrict__ dst,
                              float* __restrict__ out, int nE) {
  int t = blockIdx.x * blockDim.x + threadIdx.x;
  int e = t >> 4;
  int lm = threadIdx.x & 15;
  if (e < nE) {
    const float4* zs = (const float4*)(z + (size_t)src[e] * DIMV);
    const float4* zd = (const float4*)(z + (size_t)dst[e] * DIMV);
    float4 a = zs[lm];
    float4 b = zd[lm];
    float p = a.x * b.x + a.y * b.y + a.z * b.z + a.w * b.w;
    p += __shfl_xor(p, 8, 32);
    p += __shfl_xor(p, 4, 32);
    p += __shfl_xor(p, 2, 32);
    p += __shfl_xor(p, 1, 32);
    if (lm == 0) out[e] = p;
  }
}

// ---------------------------------------------------------------- launcher

extern "C" void kernel_launch(void* const* d_in, const int* in_sizes, int n_in,
                              void* d_out, int out_size, void* d_ws, size_t ws_size,
                              hipStream_t stream) {
  (void)n_in; (void)out_size; (void)ws_size;

  const float* x  = (const float*)d_in[0];
  const int*   ei = (const int*)d_in[1];
  const int*  pei = (const int*)d_in[2];
  const int*  nei = (const int*)d_in[3];
  const float* W1 = (const float*)d_in[4];
  const float* b1 = (const float*)d_in[5];
  const float* W2 = (const float*)d_in[6];
  const float* b2 = (const float*)d_in[7];
  float* out = (float*)d_out;

  const int N  = in_sizes[0] / DIMV;  // 100000
  const int nE = in_sizes[1] / 2;     // 1250000
  const int nP = in_sizes[2] / 2;     // 500000
  const int nN = in_sizes[3] / 2;     // 500000

  const int* src  = ei;
  const int* dst  = ei + nE;
  const int* psrc = pei;
  const int* pdst = pei + nP;
  const int* nsrc = nei;
  const int* ndst = nei + nN;

  // Workspace: deg | dinv | buf0 (25.6 MB) | buf1 (25.6 MB)  (~52 MB total)
  char* ws = (char*)d_ws;
  float* deg  = (float*)(ws);
  float* dinv = (float*)(ws + (size_t)N * 4);
  float* buf0 = (float*)(ws + (size_t)N * 8);
  float* buf1 = (float*)(ws + (size_t)N * 8 + (size_t)N * DIMV * 4);

  const int B = 256;
  const int featElems = N * DIMV;

  // Degrees + normalization
  init_deg_kernel<<<(N + B - 1) / B, B, 0, stream>>>(deg, N);
  deg_count_kernel<<<(nE + B - 1) / B, B, 0, stream>>>(dst, deg, nE);
  dinv_kernel<<<(N + B - 1) / B, B, 0, stream>>>(deg, dinv, N);

  // ---- Layer 1: h0 = x @ W1 ; agg1 = scatter(h0) ; relu(agg1 + selfloop + b1)
  gemm64_wmma_kernel<<<(N + 63) / 64, 128, 0, stream>>>(x, W1, buf0, N);
  zero4_kernel<<<(featElems / 4 + B - 1) / B, B, 0, stream>>>((float4*)buf1, featElems / 4);
  scatter_kernel<<<(nE * DIMV + B - 1) / B, B, 0, stream>>>(buf0, src, dst, dinv, buf1, nE);
  finalize_kernel<true><<<(featElems + B - 1) / B, B, 0, stream>>>(buf1, buf0, dinv, b1, N);

  // ---- Layer 2: h1 = relu_out @ W2 ; z = scatter(h1) + selfloop + b2
  gemm64_wmma_kernel<<<(N + 63) / 64, 128, 0, stream>>>(buf1, W2, buf0, N);
  zero4_kernel<<<(featElems / 4 + B - 1) / B, B, 0, stream>>>((float4*)buf1, featElems / 4);
  scatter_kernel<<<(nE * DIMV + B - 1) / B, B, 0, stream>>>(buf0, src, dst, dinv, buf1, nE);
  finalize_kernel<false><<<(featElems + B - 1) / B, B, 0, stream>>>(buf1, buf0, dinv, b2, N);

  // ---- Decode: pos scores then neg scores, concatenated in d_out
  decode_kernel<<<(nP * 16 + B - 1) / B, B, 0, stream>>>(buf1, psrc, pdst, out, nP);
  decode_kernel<<<(nN * 16 + B - 1) / B, B, 0, stream>>>(buf1, nsrc, ndst, out + nP, nN);
}